// GraphTransformerLayer_82016695484632
// MI455X (gfx1250) — compile-verified
//
#include <hip/hip_runtime.h>
#include <hip/hip_bf16.h>

typedef __attribute__((ext_vector_type(2))) float v2f;
typedef __attribute__((ext_vector_type(8))) float v8f;

#define LN_EPS 1e-5f

// ---------------------------------------------------------------------------
// Zero-fill (denominator + aggregation buffers must be zeroed every call:
// atomics accumulate and the harness does not re-poison between replays).
// ---------------------------------------------------------------------------
__global__ void gt_zero_f32(float* __restrict__ p, long long n) {
  long long i = (long long)blockIdx.x * blockDim.x + threadIdx.x;
  if (i < n) p[i] = 0.0f;
}

// ---------------------------------------------------------------------------
// Generic WMMA fp32 GEMM, shapes fixed at compile time so all B-operand
// addresses are base + immediate-offset (no per-iteration 64-bit VALU math):
//   out[M x ncols] = A[M x K_] @ W[K_ x LDW_]  (+bias) (+resid) (relu?)
//   blockIdx.x = 16-row strip, blockIdx.y = 128-col group.
//   8 waves / block, each wave owns 16 output columns.
//   V_WMMA_F32_16X16X4_F32 mapping (lane = 16*g + t):
//     A vgpr j = A[t][k+2g+j]          -> ds_load_2addr_b64 from padded LDS
//     B vgpr j = W[(k+2g+j)*LDW_ + n]  -> global_load_b32 base+imm offsets
//     D vgpr r = out[m0+r+8g][n]
// ---------------------------------------------------------------------------
template <int K_, int LDW_, int LDA_>
__global__ __launch_bounds__(256) void gt_gemm_wmma_f32(
    const float* __restrict__ A,
    const float* __restrict__ W,
    const float* __restrict__ bias,
    const float* __restrict__ resid, int ldr,
    float* __restrict__ out, int ldo,
    int M, int relu)
{
  __shared__ float Alds[16][132];            // 132-stride: bank-conflict-free
  const int m0   = blockIdx.x * 16;
  const int tid  = threadIdx.x;
  const int wave = tid >> 5;
  const int lane = tid & 31;
  const int g    = lane >> 4;
  const int t    = lane & 15;
  const int n    = blockIdx.y * 128 + wave * 16 + t;   // global output column

  // Per-lane B base pointer; every k-step load is then an immediate offset.
  const float* __restrict__ wp2 = W + (size_t)(2 * g) * LDW_ + n;

  v8f c = {0.f, 0.f, 0.f, 0.f, 0.f, 0.f, 0.f, 0.f};

#pragma unroll
  for (int kc = 0; kc < K_; kc += 128) {     // chunked so LDS stays 8.25 KB
    __syncthreads();
    // Cooperative stage of the 16 x 128 A chunk (coalesced global loads).
#pragma unroll
    for (int i = 0; i < 8; ++i) {
      int e   = tid + 256 * i;               // 0..2047
      int r   = e >> 7;
      int cc  = e & 127;
      int row = m0 + r;
      if (row >= M) row = M - 1;             // clamp (value unused if OOB row)
      Alds[r][cc] = A[(size_t)row * LDA_ + kc + cc];
    }
    __syncthreads();

#pragma unroll
    for (int k0 = 0; k0 < 128; k0 += 4) {
      v2f a;
      a.x = Alds[t][k0 + 2 * g];
      a.y = Alds[t][k0 + 2 * g + 1];
      v2f b;
      b.x = wp2[(size_t)(kc + k0) * LDW_];        // imm offset
      b.y = wp2[(size_t)(kc + k0 + 1) * LDW_];    // imm offset
      c = __builtin_amdgcn_wmma_f32_16x16x4_f32(
              false, a, false, b, (short)0, c, false, false);
    }
  }

  const float bv = bias ? bias[n] : 0.0f;
#pragma unroll
  for (int r = 0; r < 8; ++r) {
    int row = m0 + r + 8 * g;
    if (row < M) {
      float v = c[r] + bv;
      if (resid) v += resid[(size_t)row * ldr + n];
      if (relu)  v = fmaxf(v, 0.0f);
      out[(size_t)row * ldo + n] = v;        // lanes 0..15 contiguous: coalesced
    }
  }
}

// ---------------------------------------------------------------------------
// Edge phase 1: per-(edge,head) exp-score + atomic denominator scatter.
// Q/K/denom are L2-resident (102 MB + 3.2 MB << 192 MB L2).
// ---------------------------------------------------------------------------
__global__ __launch_bounds__(256) void gt_edge_scores(
    const int* __restrict__ src, const int* __restrict__ tgt,
    const float* __restrict__ Q, const float* __restrict__ K,
    float* __restrict__ scores, float* __restrict__ denom, int nEdges)
{
  int gid = blockIdx.x * 256 + threadIdx.x;
  int total = nEdges * 8;
  if (gid >= total) return;
  int e = gid >> 3;
  int h = gid & 7;
  int s = src[e];
  int t = tgt[e];
  const float4* q = (const float4*)(Q + (size_t)t * 128 + h * 16);
  const float4* k = (const float4*)(K + (size_t)s * 128 + h * 16);
  float d = 0.0f;
#pragma unroll
  for (int i = 0; i < 4; ++i) {
    float4 a = q[i], b = k[i];
    d += a.x * b.x + a.y * b.y + a.z * b.z + a.w * b.w;
  }
  float sc = __expf(d * 0.25f);              // 1/sqrt(HD) = 1/4
  scores[gid] = sc;
  unsafeAtomicAdd(&denom[(size_t)t * 8 + h], sc);   // global_atomic_add_f32
}

// ---------------------------------------------------------------------------
// Edge phase 2: alpha = score/denom[tgt], scatter alpha*V[src] into agg[tgt].
// ---------------------------------------------------------------------------
__global__ __launch_bounds__(256) void gt_edge_aggregate(
    const int* __restrict__ src, const int* __restrict__ tgt,
    const float* __restrict__ V,
    const float* __restrict__ scores, const float* __restrict__ denom,
    float* __restrict__ agg, int nEdges)
{
  int gid = blockIdx.x * 256 + threadIdx.x;
  int total = nEdges * 8;
  if (gid >= total) return;
  int e = gid >> 3;
  int h = gid & 7;
  int s = src[e];
  int t = tgt[e];
  float alpha = scores[gid] / denom[(size_t)t * 8 + h];
  const float4* vp = (const float4*)(V + (size_t)s * 128 + h * 16);
  float* o = agg + (size_t)t * 128 + h * 16;
#pragma unroll
  for (int i = 0; i < 4; ++i) {
    float4 v = vp[i];
    unsafeAtomicAdd(&o[4 * i + 0], alpha * v.x);
    unsafeAtomicAdd(&o[4 * i + 1], alpha * v.y);
    unsafeAtomicAdd(&o[4 * i + 2], alpha * v.z);
    unsafeAtomicAdd(&o[4 * i + 3], alpha * v.w);
  }
}

// ---------------------------------------------------------------------------
// LayerNorm over 128 features: one wave32 per row, 4 floats/lane,
// __shfl_xor butterfly reduction (wave32-correct on gfx1250).
// ---------------------------------------------------------------------------
__global__ __launch_bounds__(256) void gt_layernorm(
    const float* __restrict__ x,
    const float* __restrict__ gw, const float* __restrict__ bw,
    float* __restrict__ out, int nrows)
{
  int wave = threadIdx.x >> 5;
  int lane = threadIdx.x & 31;
  int row  = blockIdx.x * 8 + wave;
  if (row >= nrows) return;
  float4 v = ((const float4*)(x + (size_t)row * 128))[lane];
  float s  = v.x + v.y + v.z + v.w;
  float sq = v.x * v.x + v.y * v.y + v.z * v.z + v.w * v.w;
#pragma unroll
  for (int m = 16; m >= 1; m >>= 1) {
    s  += __shfl_xor(s, m, 32);
    sq += __shfl_xor(sq, m, 32);
  }
  float mean = s * (1.0f / 128.0f);
  float var  = sq * (1.0f / 128.0f) - mean * mean;
  float rstd = rsqrtf(var + LN_EPS);
  float4 gv = ((const float4*)gw)[lane];
  float4 bv = ((const float4*)bw)[lane];
  float4 o;
  o.x = (v.x - mean) * rstd * gv.x + bv.x;
  o.y = (v.y - mean) * rstd * gv.y + bv.y;
  o.z = (v.z - mean) * rstd * gv.z + bv.z;
  o.w = (v.w - mean) * rstd * gv.w + bv.w;
  ((float4*)(out + (size_t)row * 128))[lane] = o;
}

// ---------------------------------------------------------------------------
extern "C" void kernel_launch(void* const* d_in, const int* in_sizes, int n_in,
                              void* d_out, int out_size, void* d_ws, size_t ws_size,
                              hipStream_t stream)
{
  const float* node_feat  = (const float*)d_in[0];
  const int*   edge_index = (const int*)  d_in[1];
  const float* Wq    = (const float*)d_in[2];
  const float* Wk    = (const float*)d_in[3];
  const float* Wv    = (const float*)d_in[4];
  const float* Wo    = (const float*)d_in[5];
  const float* bo    = (const float*)d_in[6];
  const float* ln1_g = (const float*)d_in[7];
  const float* ln1_b = (const float*)d_in[8];
  const float* W1    = (const float*)d_in[9];
  const float* b1    = (const float*)d_in[10];
  const float* W2    = (const float*)d_in[11];
  const float* b2    = (const float*)d_in[12];
  const float* ln2_g = (const float*)d_in[13];
  const float* ln2_b = (const float*)d_in[14];

  const int D = 128;
  const int H = 8;
  const int N = in_sizes[0] / D;      // 100000
  const int E = in_sizes[1] / 2;      // 1600000
  const int* src = edge_index;        // edge_index[0, :]
  const int* tgt = edge_index + E;    // edge_index[1, :]

  // Workspace layout (floats), with buffer reuse:
  //   Q | K | V | scores(->tmp1) | denom | agg(->x1) | H1 ; tmp2 reuses Q.
  float* ws = (float*)d_ws;
  const size_t ND = (size_t)N * D;
  const size_t NH = (size_t)N * H;
  const size_t EH = (size_t)E * H;
  const size_t sc_sz = (EH > ND) ? EH : ND;
  float* Q      = ws;
  float* Kb     = Q + ND;
  float* V      = Kb + ND;
  float* scores = V + ND;            // later reused as tmp1 (pre-LN1)
  float* denom  = scores + sc_sz;
  float* agg    = denom + NH;        // later reused as x1 (post-LN1)
  float* H1     = agg + ND;          // FFN hidden [N, 2D]
  float* tmp2   = Q;                 // pre-LN2, reuses Q region

  const int gm = (N + 15) / 16;      // 16-row GEMM strips
  const int ge = (int)((EH + 255) / 256);
  const int gl = (N + 7) / 8;

  // 0) zero denom + agg (contiguous region)
  {
    long long z = (long long)(NH + ND);
    gt_zero_f32<<<(int)((z + 255) / 256), 256, 0, stream>>>(denom, z);
  }

  // 1) Q, K, V projections   (K=128, ldw=128, lda=128)
  gt_gemm_wmma_f32<128, 128, 128><<<dim3(gm, 1), 256, 0, stream>>>(
      node_feat, Wq, nullptr, nullptr, 0, Q,  D, N, 0);
  gt_gemm_wmma_f32<128, 128, 128><<<dim3(gm, 1), 256, 0, stream>>>(
      node_feat, Wk, nullptr, nullptr, 0, Kb, D, N, 0);
  gt_gemm_wmma_f32<128, 128, 128><<<dim3(gm, 1), 256, 0, stream>>>(
      node_feat, Wv, nullptr, nullptr, 0, V,  D, N, 0);

  // 2) per-edge exp-scores + softmax denominators
  gt_edge_scores<<<ge, 256, 0, stream>>>(src, tgt, Q, Kb, scores, denom, E);

  // 3) alpha-weighted V aggregation
  gt_edge_aggregate<<<ge, 256, 0, stream>>>(src, tgt, V, scores, denom, agg, E);

  // 4) output projection + bias + residual -> tmp1 (scores region); LN1 -> x1 (agg region)
  gt_gemm_wmma_f32<128, 128, 128><<<dim3(gm, 1), 256, 0, stream>>>(
      agg, Wo, bo, node_feat, D, scores, D, N, 0);
  gt_layernorm<<<gl, 256, 0, stream>>>(scores, ln1_g, ln1_b, agg, N);

  // 5) FFN: relu(x1 @ W1 + b1) -> H1 ; H1 @ W2 + b2 + x1 -> tmp2 ; LN2 -> out
  gt_gemm_wmma_f32<128, 256, 128><<<dim3(gm, 2), 256, 0, stream>>>(
      agg, W1, b1, nullptr, 0, H1, 2 * D, N, 1);
  gt_gemm_wmma_f32<256, 128, 256><<<dim3(gm, 1), 256, 0, stream>>>(
      H1, W2, b2, agg, D, tmp2, D, N, 0);
  gt_layernorm<<<gl, 256, 0, stream>>>(tmp2, ln2_g, ln2_b, (float*)d_out, N);
}